// GraphDecoder_65747359367209
// MI455X (gfx1250) — compile-verified
//
#include <hip/hip_runtime.h>
#include <cstddef>

typedef __attribute__((ext_vector_type(2))) float v2f;
typedef __attribute__((ext_vector_type(4))) float v4f;
typedef __attribute__((ext_vector_type(8))) float v8f;

#define HID   128
#define LATD  256
#define NNODE 32768
#define EDGES 1048576

__device__ __forceinline__ float silu_f(float v) {
  return v / (1.0f + __expf(-v));
}

// ---------------------------------------------------------------- utilities
__global__ void k_zero(float* __restrict__ p, size_t n) {
  size_t i = (size_t)blockIdx.x * blockDim.x + threadIdx.x;
  if (i < n) p[i] = 0.0f;
}

// a = silu(z @ Wl1 + bl1); z:(256), Wl1:(256,256)
__global__ void k_latent(const float* __restrict__ z, const float* __restrict__ Wl1,
                         const float* __restrict__ bl1, float* __restrict__ a) {
  __shared__ float zs[LATD];
  int t = threadIdx.x;
  zs[t] = z[t];
  __syncthreads();
  float acc = bl1[t];
  for (int k = 0; k < LATD; ++k) acc += zs[k] * Wl1[k * 256 + t];
  a[t] = silu_f(acc);
}

// h0[col] = a @ Wl2[:,col] + bl2[col]; Wl2:(256, 524288).
// Pure 512MB weight stream: non-temporal b128 loads, 4 cols / thread.
__global__ void k_wl2(const float* __restrict__ a, const float* __restrict__ Wl2,
                      const float* __restrict__ bl2, float* __restrict__ h0) {
  __shared__ float as[LATD];
  int t = threadIdx.x;
  as[t] = a[t];
  __syncthreads();
  size_t col = ((size_t)blockIdx.x * 256 + t) * 4;
  v4f acc = *(const v4f*)(bl2 + col);
  for (int k = 0; k < LATD; ++k) {
    v4f w = __builtin_nontemporal_load((const v4f*)(Wl2 + (size_t)k * 524288 + col));
    float av = as[k];
    acc.x += av * w.x;
    acc.y += av * w.y;
    acc.z += av * w.z;
    acc.w += av * w.w;
  }
  *(v4f*)(h0 + col) = acc;
}

// dst[keep[i]] = src[i] (dst pre-zeroed). grid = Mprev, block = 128
__global__ void k_unpool(const float* __restrict__ src, const int* __restrict__ keep,
                         float* __restrict__ dst) {
  int i = blockIdx.x, c = threadIdx.x;
  dst[(size_t)keep[i] * HID + c] = src[(size_t)i * HID + c];
}

// Row LayerNorm: one wave32 per row, 4 channels per lane. grid = M/8, block 256
__global__ void k_ln(const float* __restrict__ x, const float* __restrict__ g,
                     const float* __restrict__ b, float* __restrict__ y, int M) {
  int row = blockIdx.x * 8 + (threadIdx.x >> 5);
  int lane = threadIdx.x & 31;
  const float4 v = *(const float4*)(x + (size_t)row * HID + lane * 4);
  float s = v.x + v.y + v.z + v.w;
  for (int o = 16; o; o >>= 1) s += __shfl_xor(s, o);
  float mean = s * (1.0f / 128.0f);
  float dx = v.x - mean, dy = v.y - mean, dz = v.z - mean, dw = v.w - mean;
  float q = dx * dx + dy * dy + dz * dz + dw * dw;
  for (int o = 16; o; o >>= 1) q += __shfl_xor(q, o);
  float rs = rsqrtf(q * (1.0f / 128.0f) + 1e-5f);
  const float4 gg = *(const float4*)(g + lane * 4);
  const float4 bb = *(const float4*)(b + lane * 4);
  float4 out;
  out.x = dx * rs * gg.x + bb.x;
  out.y = dy * rs * gg.y + bb.y;
  out.z = dz * rs * gg.z + bb.z;
  out.w = dw * rs * gg.w + bb.w;
  *(float4*)(y + (size_t)row * HID + lane * 4) = out;
}

// One wave per edge; skip edges outside the subgraph (mask==0 contributes 0).
__global__ void k_edge(const int* __restrict__ er, const int* __restrict__ ec,
                       const float* __restrict__ hn, float* __restrict__ nsum, int Np) {
  int e = blockIdx.x * 8 + (threadIdx.x >> 5);
  int lane = threadIdx.x & 31;
  int r = er[e], c = ec[e];
  if (r >= Np || c >= Np) return;
  const float4 v = *(const float4*)(hn + (size_t)c * HID + lane * 4);
  float* dst = nsum + (size_t)r * HID + lane * 4;
  atomicAdd(dst + 0, v.x);
  atomicAdd(dst + 1, v.y);
  atomicAdd(dst + 2, v.z);
  atomicAdd(dst + 3, v.w);
}

// Y = [R +] act(X @ W + bias). 256 thr = 8 waves; block tile 32 rows x 128 cols.
// Each wave: two 16x16 output tiles sharing the B fragment (2 indep acc chains).
__global__ void k_gemm(const float* __restrict__ X, const float* __restrict__ W,
                       const float* __restrict__ bias, const float* __restrict__ R,
                       int do_silu, float* __restrict__ Y) {
  __shared__ float sX[32][132];
  int row0 = blockIdx.x * 32;
  int tid = threadIdx.x;
  for (int i = tid; i < 32 * HID; i += 256) {
    int r = i >> 7, c = i & 127;
    sX[r][c] = X[(size_t)(row0 + r) * HID + c];
  }
  __syncthreads();
  int wave = tid >> 5, lane = tid & 31;
  int col0 = wave * 16, n = lane & 15, hh = lane >> 4, m = lane & 15;
  v8f acc0 = {}, acc1 = {};
  for (int kk = 0; kk < 32; ++kk) {
    int k0 = kk * 4 + hh * 2;
    v2f b, a0, a1;
    b.x = W[(size_t)k0 * HID + col0 + n];
    b.y = W[(size_t)(k0 + 1) * HID + col0 + n];
    a0.x = sX[m][k0];
    a0.y = sX[m][k0 + 1];
    a1.x = sX[m + 16][k0];
    a1.y = sX[m + 16][k0 + 1];
    acc0 = __builtin_amdgcn_wmma_f32_16x16x4_f32(false, a0, false, b, (short)0, acc0,
                                                 false, false);
    acc1 = __builtin_amdgcn_wmma_f32_16x16x4_f32(false, a1, false, b, (short)0, acc1,
                                                 false, false);
  }
  float bv = bias[col0 + n];
  for (int r = 0; r < 8; ++r) {
    int orow = row0 + r + 8 * hh;
    float v0 = acc0[r] + bv;
    float v1 = acc1[r] + bv;
    if (do_silu) { v0 = silu_f(v0); v1 = silu_f(v1); }
    if (R) {
      v0 += R[(size_t)orow * HID + col0 + n];
      v1 += R[(size_t)(orow + 16) * HID + col0 + n];
    }
    Y[(size_t)orow * HID + col0 + n] = v0;
    Y[(size_t)(orow + 16) * HID + col0 + n] = v1;
  }
}

// Fused GIN block tail: t=(1+eps)*hn+nsum; y = x + silu(t@W1+b1)@W2 + b2
// Block tile 32 rows x 128 cols; 8 waves, 2 tiles per wave sharing B fragments.
__global__ void k_mlp(const float* __restrict__ x, const float* __restrict__ hn,
                      const float* __restrict__ ns, const float* __restrict__ W1,
                      const float* __restrict__ b1, const float* __restrict__ W2,
                      const float* __restrict__ b2, const float* __restrict__ epsv,
                      float* __restrict__ y) {
  __shared__ float sT[32][132];
  __shared__ float sM[32][132];
  int row0 = blockIdx.x * 32;
  int tid = threadIdx.x;
  float ep = 1.0f + epsv[0];
  for (int i = tid; i < 32 * HID; i += 256) {
    int r = i >> 7, c = i & 127;
    size_t g = (size_t)(row0 + r) * HID + c;
    sT[r][c] = ep * hn[g] + ns[g];
  }
  __syncthreads();
  int wave = tid >> 5, lane = tid & 31;
  int col0 = wave * 16, n = lane & 15, hh = lane >> 4, m = lane & 15;
  v8f acc0 = {}, acc1 = {};
  for (int kk = 0; kk < 32; ++kk) {
    int k0 = kk * 4 + hh * 2;
    v2f b, a0, a1;
    b.x = W1[(size_t)k0 * HID + col0 + n];
    b.y = W1[(size_t)(k0 + 1) * HID + col0 + n];
    a0.x = sT[m][k0];
    a0.y = sT[m][k0 + 1];
    a1.x = sT[m + 16][k0];
    a1.y = sT[m + 16][k0 + 1];
    acc0 = __builtin_amdgcn_wmma_f32_16x16x4_f32(false, a0, false, b, (short)0, acc0,
                                                 false, false);
    acc1 = __builtin_amdgcn_wmma_f32_16x16x4_f32(false, a1, false, b, (short)0, acc1,
                                                 false, false);
  }
  float bv = b1[col0 + n];
  for (int r = 0; r < 8; ++r) {
    sM[r + 8 * hh][col0 + n] = silu_f(acc0[r] + bv);
    sM[r + 8 * hh + 16][col0 + n] = silu_f(acc1[r] + bv);
  }
  __syncthreads();
  v8f acc2 = {}, acc3 = {};
  for (int kk = 0; kk < 32; ++kk) {
    int k0 = kk * 4 + hh * 2;
    v2f b, a0, a1;
    b.x = W2[(size_t)k0 * HID + col0 + n];
    b.y = W2[(size_t)(k0 + 1) * HID + col0 + n];
    a0.x = sM[m][k0];
    a0.y = sM[m][k0 + 1];
    a1.x = sM[m + 16][k0];
    a1.y = sM[m + 16][k0 + 1];
    acc2 = __builtin_amdgcn_wmma_f32_16x16x4_f32(false, a0, false, b, (short)0, acc2,
                                                 false, false);
    acc3 = __builtin_amdgcn_wmma_f32_16x16x4_f32(false, a1, false, b, (short)0, acc3,
                                                 false, false);
  }
  float bv2 = b2[col0 + n];
  for (int r = 0; r < 8; ++r) {
    int orow = row0 + r + 8 * hh;
    y[(size_t)orow * HID + col0 + n] =
        x[(size_t)orow * HID + col0 + n] + acc2[r] + bv2;
    y[(size_t)(orow + 16) * HID + col0 + n] =
        x[(size_t)(orow + 16) * HID + col0 + n] + acc3[r] + bv2;
  }
}

// pos mean/std (ddof=1) per dim -> stats[0..2]=mean, stats[3..5]=std+1e-8
__global__ void k_posstats(const float* __restrict__ pos, float* __restrict__ stats) {
  __shared__ float rs[256], rq[256];
  int t = threadIdx.x;
  for (int d = 0; d < 3; ++d) {
    float s = 0.0f, q = 0.0f;
    for (int i = t; i < NNODE; i += 256) {
      float v = pos[(size_t)i * 3 + d];
      s += v;
      q += v * v;
    }
    rs[t] = s;
    rq[t] = q;
    __syncthreads();
    for (int o = 128; o; o >>= 1) {
      if (t < o) { rs[t] += rs[t + o]; rq[t] += rq[t + o]; }
      __syncthreads();
    }
    if (t == 0) {
      float mean = rs[0] / (float)NNODE;
      float var = (rq[0] - (float)NNODE * mean * mean) / (float)(NNODE - 1);
      stats[d] = mean;
      stats[3 + d] = sqrtf(var) + 1e-8f;
    }
    __syncthreads();
  }
}

// t[row,c] = silu(pn @ Wp1 + bp1), pn = (pos-mean)/(std+1e-8)
__global__ void k_posembed(const float* __restrict__ pos, const float* __restrict__ stats,
                           const float* __restrict__ Wp1, const float* __restrict__ bp1,
                           float* __restrict__ t) {
  size_t idx = (size_t)blockIdx.x * 256 + threadIdx.x;
  int row = (int)(idx >> 7), c = (int)(idx & 127);
  float p0 = (pos[(size_t)row * 3 + 0] - stats[0]) / stats[3];
  float p1 = (pos[(size_t)row * 3 + 1] - stats[1]) / stats[4];
  float p2 = (pos[(size_t)row * 3 + 2] - stats[2]) / stats[5];
  float v = bp1[c] + p0 * Wp1[c] + p1 * Wp1[128 + c] + p2 * Wp1[256 + c];
  t[idx] = silu_f(v);
}

// out = silu(LN(h)*on_g+on_b) @ Wo + bo ; one wave per row
__global__ void k_out(const float* __restrict__ h, const float* __restrict__ g,
                      const float* __restrict__ b, const float* __restrict__ Wo,
                      const float* __restrict__ bo, float* __restrict__ out) {
  int row = blockIdx.x * 8 + (threadIdx.x >> 5);
  int lane = threadIdx.x & 31;
  const float4 v = *(const float4*)(h + (size_t)row * HID + lane * 4);
  float s = v.x + v.y + v.z + v.w;
  for (int o = 16; o; o >>= 1) s += __shfl_xor(s, o);
  float mean = s * (1.0f / 128.0f);
  float dx = v.x - mean, dy = v.y - mean, dz = v.z - mean, dw = v.w - mean;
  float q = dx * dx + dy * dy + dz * dz + dw * dw;
  for (int o = 16; o; o >>= 1) q += __shfl_xor(q, o);
  float rs = rsqrtf(q * (1.0f / 128.0f) + 1e-5f);
  const float4 gg = *(const float4*)(g + lane * 4);
  const float4 bb = *(const float4*)(b + lane * 4);
  float t0 = silu_f(dx * rs * gg.x + bb.x);
  float t1 = silu_f(dy * rs * gg.y + bb.y);
  float t2 = silu_f(dz * rs * gg.z + bb.z);
  float t3 = silu_f(dw * rs * gg.w + bb.w);
  int c = lane * 4;
  float o0 = t0 * Wo[c * 3 + 0] + t1 * Wo[(c + 1) * 3 + 0] +
             t2 * Wo[(c + 2) * 3 + 0] + t3 * Wo[(c + 3) * 3 + 0];
  float o1 = t0 * Wo[c * 3 + 1] + t1 * Wo[(c + 1) * 3 + 1] +
             t2 * Wo[(c + 2) * 3 + 1] + t3 * Wo[(c + 3) * 3 + 1];
  float o2 = t0 * Wo[c * 3 + 2] + t1 * Wo[(c + 1) * 3 + 2] +
             t2 * Wo[(c + 2) * 3 + 2] + t3 * Wo[(c + 3) * 3 + 2];
  for (int o = 16; o; o >>= 1) {
    o0 += __shfl_xor(o0, o);
    o1 += __shfl_xor(o1, o);
    o2 += __shfl_xor(o2, o);
  }
  if (lane == 0) {
    out[(size_t)row * 3 + 0] = o0 + bo[0];
    out[(size_t)row * 3 + 1] = o1 + bo[1];
    out[(size_t)row * 3 + 2] = o2 + bo[2];
  }
}

// ---------------------------------------------------------------- host
extern "C" void kernel_launch(void* const* d_in, const int* in_sizes, int n_in,
                              void* d_out, int out_size, void* d_ws, size_t ws_size,
                              hipStream_t stream) {
  (void)in_sizes; (void)n_in; (void)out_size; (void)ws_size;
  const float* z    = (const float*)d_in[0];
  const float* pos  = (const float*)d_in[1];
  const int*   er   = (const int*)d_in[2];
  const int*   ec   = (const int*)d_in[3];
  const int*   keep0 = (const int*)d_in[4];   // 16384 -> level 32768
  const int*   keep1 = (const int*)d_in[5];   // 8192  -> level 16384
  const int*   keep2 = (const int*)d_in[6];   // 4096  -> level 8192
  const float* Wl1 = (const float*)d_in[7];
  const float* bl1 = (const float*)d_in[8];
  const float* Wl2 = (const float*)d_in[9];
  const float* bl2 = (const float*)d_in[10];
  const float* Wp1 = (const float*)d_in[11];
  const float* bp1 = (const float*)d_in[12];
  const float* Wp2 = (const float*)d_in[13];
  const float* bp2 = (const float*)d_in[14];
  const float* Wu  = (const float*)d_in[15];
  const float* bu  = (const float*)d_in[16];
  const float* lng = (const float*)d_in[17];
  const float* lnb = (const float*)d_in[18];
  const float* W1  = (const float*)d_in[19];
  const float* b1  = (const float*)d_in[20];
  const float* W2  = (const float*)d_in[21];
  const float* b2  = (const float*)d_in[22];
  const float* epsp = (const float*)d_in[23];
  const float* ong = (const float*)d_in[24];
  const float* onb = (const float*)d_in[25];
  const float* Wo  = (const float*)d_in[26];
  const float* bo  = (const float*)d_in[27];

  float* wsf   = (float*)d_ws;
  float* a_vec = wsf;                                // 256
  float* stats = wsf + 256;                          // 6
  float* bufA  = wsf + 512;                          // 32768*128
  float* bufB  = bufA + (size_t)NNODE * HID;
  float* hnb   = bufB + (size_t)NNODE * HID;
  float* nsb   = hnb + (size_t)NNODE * HID;

  k_latent<<<1, 256, 0, stream>>>(z, Wl1, bl1, a_vec);
  k_wl2<<<512, 256, 0, stream>>>(a_vec, Wl2, bl2, bufA);  // h: (4096,128)

  const int levels[4] = {4096, 8192, 16384, 32768};
  const int* keeps[3] = {keep2, keep1, keep0};
  float* x = bufA;
  float* y = bufB;
  int blk = 0;

  for (int d = 0; d < 3; ++d) {
    int Mp = levels[d], Np = levels[d + 1];
    size_t nel = (size_t)Np * HID;
    k_zero<<<(unsigned)((nel + 255) / 256), 256, 0, stream>>>(y, nel);
    k_unpool<<<Mp, 128, 0, stream>>>(x, keeps[d], y);
    k_gemm<<<Np / 32, 256, 0, stream>>>(y, Wu + (size_t)d * HID * HID, bu + d * HID,
                                        nullptr, 0, x);  // x = unpooled @ Wu + bu
    for (int bidx = 0; bidx < 2; ++bidx, ++blk) {
      k_ln<<<Np / 8, 256, 0, stream>>>(x, lng + blk * HID, lnb + blk * HID, hnb, Np);
      k_zero<<<(unsigned)((nel + 255) / 256), 256, 0, stream>>>(nsb, nel);
      k_edge<<<EDGES / 8, 256, 0, stream>>>(er, ec, hnb, nsb, Np);
      k_mlp<<<Np / 32, 256, 0, stream>>>(x, hnb, nsb, W1 + (size_t)blk * HID * HID,
                                         b1 + blk * HID, W2 + (size_t)blk * HID * HID,
                                         b2 + blk * HID, epsp + blk, y);
      float* t = x; x = y; y = t;
    }
  }

  // position embedding: h += silu(pn@Wp1+bp1) @ Wp2 + bp2
  k_posstats<<<1, 256, 0, stream>>>(pos, stats);
  k_posembed<<<(NNODE * HID) / 256, 256, 0, stream>>>(pos, stats, Wp1, bp1, hnb);
  k_gemm<<<NNODE / 32, 256, 0, stream>>>(hnb, Wp2, bp2, x, 0, y);
  { float* t = x; x = y; y = t; }

  // final two blocks on the full graph (Np == N, all edges valid)
  for (int bidx = 0; bidx < 2; ++bidx, ++blk) {
    size_t nel = (size_t)NNODE * HID;
    k_ln<<<NNODE / 8, 256, 0, stream>>>(x, lng + blk * HID, lnb + blk * HID, hnb, NNODE);
    k_zero<<<(unsigned)((nel + 255) / 256), 256, 0, stream>>>(nsb, nel);
    k_edge<<<EDGES / 8, 256, 0, stream>>>(er, ec, hnb, nsb, NNODE);
    k_mlp<<<NNODE / 32, 256, 0, stream>>>(x, hnb, nsb, W1 + (size_t)blk * HID * HID,
                                          b1 + blk * HID, W2 + (size_t)blk * HID * HID,
                                          b2 + blk * HID, epsp + blk, y);
    float* t = x; x = y; y = t;
  }

  k_out<<<NNODE / 8, 256, 0, stream>>>(x, ong, onb, Wo, bo, (float*)d_out);
}